// VolumeRenderer_76922864271741
// MI455X (gfx1250) — compile-verified
//
#include <hip/hip_runtime.h>
#include <stdint.h>

#define NSAMP 128
#define CHUNK 32
#define NCHUNK 4
#define WAVES_PER_BLOCK 8

// ---- CDNA5 async global->LDS helpers (ASYNCcnt-tracked DMA path) ----------

__device__ __forceinline__ uint32_t lds_off32(const void* p) {
    // generic pointer into LDS aperture: low 32 bits are the workgroup-relative
    // LDS byte offset
    return (uint32_t)(uintptr_t)p;
}

__device__ __forceinline__ void async_ld_b32(uint32_t lds_dst, const void* gsrc) {
    asm volatile("global_load_async_to_lds_b32 %0, %1, off"
                 :: "v"(lds_dst), "v"((uint64_t)(uintptr_t)gsrc)
                 : "memory");
}

__device__ __forceinline__ void async_ld_b128(uint32_t lds_dst, const void* gsrc) {
    asm volatile("global_load_async_to_lds_b128 %0, %1, off"
                 :: "v"(lds_dst), "v"((uint64_t)(uintptr_t)gsrc)
                 : "memory");
}

__device__ __forceinline__ void wait_async_le3() {
    asm volatile("s_wait_asynccnt 0x3" ::: "memory");
}
__device__ __forceinline__ void wait_async_0() {
    asm volatile("s_wait_asynccnt 0x0" ::: "memory");
}
__device__ __forceinline__ void wait_ds_0() {
    asm volatile("s_wait_dscnt 0x0" ::: "memory");
}

// ---------------------------------------------------------------------------

__global__ __launch_bounds__(WAVES_PER_BLOCK * 32)
void volume_render_kernel(const float* __restrict__ density,
                          const float* __restrict__ feature,
                          const float* __restrict__ depth,
                          float* __restrict__ out,
                          int M) {
    // double-buffered per-wave staging buffers (10 KB / 256-thread block)
    __shared__ __align__(16) float ldsD[WAVES_PER_BLOCK][2][CHUNK];
    __shared__ __align__(16) float ldsZ[WAVES_PER_BLOCK][2][CHUNK];
    __shared__ __align__(16) float ldsF[WAVES_PER_BLOCK][2][CHUNK * 3];

    const int lane = threadIdx.x & 31;
    const int wv   = threadIdx.x >> 5;
    const int ray  = blockIdx.x * WAVES_PER_BLOCK + wv;   // uniform per wave
    if (ray >= M) return;

    const float* dRay = density + (size_t)ray * NSAMP;
    const float* zRay = depth   + (size_t)ray * NSAMP;
    const float* fRay = feature + (size_t)ray * (NSAMP * 3);

    // issue one chunk's staging: 3 async DMA instructions, all fully coalesced
    auto issue = [&](int c) {
        const int buf = c & 1;
        async_ld_b32(lds_off32(&ldsD[wv][buf][lane]), dRay + c * CHUNK + lane);
        async_ld_b32(lds_off32(&ldsZ[wv][buf][lane]), zRay + c * CHUNK + lane);
        if (lane < 24)  // 24 lanes x 16B = 384B = 32 samples x 3 floats
            async_ld_b128(lds_off32(&ldsF[wv][buf][lane * 4]),
                          fRay + c * (CHUNK * 3) + lane * 4);
    };

    issue(0);
    issue(1);

    float carry = 0.0f;                 // prefix carried across chunks
    float acc0 = 0.f, acc1 = 0.f, acc2 = 0.f, acc3 = 0.f;
    const float L2E = 1.4426950408889634f;

    #pragma unroll
    for (int c = 0; c < NCHUNK; ++c) {
        const int buf = c & 1;
        const int n   = c * CHUNK + lane;

        // lane-31 boundary depth (first sample of next chunk) — one masked dword
        float z_tail = 0.0f;
        if (lane == 31 && n < NSAMP - 1) z_tail = zRay[n + 1];

        // chunk c staged and visible
        if (c < NCHUNK - 1) wait_async_le3();
        else                wait_async_0();

        float sigma = ldsD[wv][buf][lane];
        float zn    = ldsZ[wv][buf][lane];
        float f0    = ldsF[wv][buf][lane * 3 + 0];
        float f1    = ldsF[wv][buf][lane * 3 + 1];
        float f2    = ldsF[wv][buf][lane * 3 + 2];

        // make sure DS reads retired before the async engine refills this buffer
        wait_ds_0();
        if (c + 2 < NCHUNK) issue(c + 2);

        // delta_n = z[n+1] - z[n];  terminal delta = 1e10
        float znext = __shfl_down(zn, 1, 32);
        if (lane == 31) znext = z_tail;
        float delta = (n == NSAMP - 1) ? 1e10f : (znext - zn);
        float tau   = sigma * delta;

        // wave32 inclusive scan of tau (5 shuffle-add steps)
        float s = tau;
        #pragma unroll
        for (int off = 1; off < 32; off <<= 1) {
            float y = __shfl_up(s, off, 32);
            if (lane >= off) s += y;
        }

        float excl = carry + (s - tau);                       // exclusive prefix
        float T    = __builtin_amdgcn_exp2f(-excl * L2E);     // transmittance
        float e    = __builtin_amdgcn_exp2f(-tau * L2E);
        float w    = T - T * e;                               // T*(1-exp(-tau))

        acc0 += w * f0;
        acc1 += w * f1;
        acc2 += w * f2;
        acc3 += w * zn;

        carry += __shfl(s, 31, 32);                           // chunk total
    }

    // wave butterfly reduction of the 4 accumulators
    #pragma unroll
    for (int off = 16; off > 0; off >>= 1) {
        acc0 += __shfl_xor(acc0, off, 32);
        acc1 += __shfl_xor(acc1, off, 32);
        acc2 += __shfl_xor(acc2, off, 32);
        acc3 += __shfl_xor(acc3, off, 32);
    }

    if (lane == 0) {
        float4 r = make_float4(acc0, acc1, acc2, acc3);
        *reinterpret_cast<float4*>(out + (size_t)ray * 4) = r;
    }
}

extern "C" void kernel_launch(void* const* d_in, const int* in_sizes, int n_in,
                              void* d_out, int out_size, void* d_ws, size_t ws_size,
                              hipStream_t stream) {
    const float* density = (const float*)d_in[0];  // [M,128,1]
    const float* feature = (const float*)d_in[1];  // [M,128,3]
    const float* depth   = (const float*)d_in[2];  // [M,128]
    float* out = (float*)d_out;                    // [M,4]

    const int M = in_sizes[0] / NSAMP;
    const int blocks = (M + WAVES_PER_BLOCK - 1) / WAVES_PER_BLOCK;

    hipLaunchKernelGGL(volume_render_kernel,
                       dim3(blocks), dim3(WAVES_PER_BLOCK * 32), 0, stream,
                       density, feature, depth, out, M);
}